// GraphResnetWithSampling_15032385536064
// MI455X (gfx1250) — compile-verified
//
#include <hip/hip_runtime.h>

// GNN forward: proj GEMM -> edge scatter -> fused MLP0 -> 2x edge scatter -> fused MLP1.
// GEMMs use V_WMMA_F32_16X16X4_F32; weights staged to LDS with TENSOR_LOAD_TO_LDS (TDM)
// using descriptor-side padding to produce a bank-conflict-free [rows][65] layout.

#define FDIM 64

typedef __attribute__((ext_vector_type(2))) float v2f;
typedef __attribute__((ext_vector_type(8))) float v8f;

__device__ __forceinline__ float lrelu(float x) { return x > 0.f ? x : 0.1f * x; }

__device__ __forceinline__ v8f wmma4(v2f a, v2f b, v8f c) {
  return __builtin_amdgcn_wmma_f32_16x16x4_f32(false, a, false, b, (short)0, c,
                                               false, false);
}

#if __has_builtin(__builtin_amdgcn_tensor_load_to_lds) && \
    __has_builtin(__builtin_amdgcn_s_wait_tensorcnt)
#define HAVE_TDM 1
#else
#define HAVE_TDM 0
#endif

#if HAVE_TDM
typedef __attribute__((ext_vector_type(4))) unsigned int u32x4;
typedef __attribute__((ext_vector_type(8))) int i32x8;
typedef __attribute__((ext_vector_type(4))) int i32x4;

// 2D row-major tensor (nrows x ncols, fp32) -> LDS with +1 dword pad per 64
// dwords (pad_interval=5, pad_amount=0), i.e. an LDS layout of [nrows][65]
// when ncols==64. ncols must be a multiple of 64 here (we only use 64).
__device__ __forceinline__ void tdm_load_2d_padded(const void* gsrc,
                                                   void* lds_dst,
                                                   unsigned ncols,
                                                   unsigned nrows) {
  const unsigned long long ga = (unsigned long long)gsrc;
  u32x4 g0;
  g0[0] = 1u;  // count=1 (valid descriptor), user mode, no gather
  g0[1] = (unsigned)((unsigned long long)lds_dst & 0xffffffffu);  // lds_addr
  g0[2] = (unsigned)(ga & 0xffffffffu);                  // global_addr[31:0]
  g0[3] = (unsigned)((ga >> 32) & 0x01ffffffu)           // global_addr[56:32]
          | 0x80000000u;                                 // type=2 ("image")
  i32x8 g1;
  g1[0] = (2 << 16)      // data_size = 4 bytes
          | (1 << 20)    // pad_enable
          | (5 << 22);   // pad_interval: every 64 dwords  (pad_amount=0 -> 1 dw)
  g1[1] = (int)((ncols & 0xffffu) << 16);                          // dim0 lo
  g1[2] = (int)(((ncols >> 16) & 0xffffu) | ((nrows & 0xffffu) << 16));
  g1[3] = (int)(((nrows >> 16) & 0xffffu) | ((ncols & 0xffffu) << 16));  // tile0
  g1[4] = (int)(nrows & 0xffffu);  // tile_dim1 ; tile_dim2 = 0
  g1[5] = (int)ncols;              // tensor_dim0_stride lo
  g1[6] = 0;
  g1[7] = 0;
  i32x4 z4 = {};
#if defined(__clang_major__) && __clang_major__ >= 23
  i32x8 z8 = {};
  __builtin_amdgcn_tensor_load_to_lds(g0, g1, z4, z4, z8, 0);
#else
  __builtin_amdgcn_tensor_load_to_lds(g0, g1, z4, z4, 0);
#endif
}
#endif  // HAVE_TDM

// ---------------------------------------------------------------- utilities
__global__ void zero_kernel(float* __restrict__ p, int count) {
  for (int i = blockIdx.x * blockDim.x + threadIdx.x; i < count;
       i += gridDim.x * blockDim.x)
    p[i] = 0.f;
}

__global__ void deg_kernel(const int* __restrict__ dst, float* __restrict__ deg,
                           int e) {
  int i = blockIdx.x * blockDim.x + threadIdx.x;
  if (i < e) atomicAdd(&deg[dst[i]], 1.0f);
}

// out[dst[e]] += feat[src[e]] ; one thread per (edge, float4-chunk)
__global__ void scatter_kernel(const float* __restrict__ feat,
                               const int* __restrict__ src,
                               const int* __restrict__ dst,
                               float* __restrict__ out, int e) {
  int i = blockIdx.x * blockDim.x + threadIdx.x;
  if (i >= e * 16) return;
  int eidx = i >> 4;
  int c4 = (i & 15) * 4;
  int s = src[eidx];
  int d = dst[eidx];
  const float4 v = *(const float4*)&feat[(size_t)s * FDIM + c4];
  float* o = &out[(size_t)d * FDIM + c4];
  atomicAdd(o + 0, v.x);
  atomicAdd(o + 1, v.y);
  atomicAdd(o + 2, v.z);
  atomicAdd(o + 3, v.w);
}

// ------------------------------------------------- proj: h0 = emb + lrelu(content@W+b)
// 32 rows/block, 8 waves, each wave owns one 16x16 output tile.
// content tile staged in LDS (padded), proj_w staged via TDM (padded by the DMA).
__global__ __launch_bounds__(256) void proj_kernel(
    const float* __restrict__ content, const float* __restrict__ node_emb,
    const float* __restrict__ pw, const float* __restrict__ pb,
    const int* __restrict__ nid, float* __restrict__ h0, int n) {
  __shared__ float X[32][257];  // 32x256 content tile, +1 pad vs 64 banks
  __shared__ float W[256][65];  // 256x64 proj_w, padded
  const int tid = threadIdx.x;
  const int base = blockIdx.x * 32;

#if HAVE_TDM
  if (tid < 32) tdm_load_2d_padded(pw, &W[0][0], 64, 256);
#endif
  for (int i = tid; i < 32 * 256; i += 256) {
    const int row = i >> 8, col = i & 255;
    const int grow = min(base + row, n - 1);
    X[row][col] = content[(size_t)grow * 256 + col];
  }
#if HAVE_TDM
  if (tid < 32) __builtin_amdgcn_s_wait_tensorcnt(0);
#else
  for (int i = tid; i < 256 * 64; i += 256) W[i >> 6][i & 63] = pw[i];
#endif
  __syncthreads();

  const int lane = tid & 31;
  const int wave = tid >> 5;
  const int m = lane & 15;
  const int kh = lane >> 4;
  const int r0 = (wave >> 2) * 16;  // local row-tile base (0 or 16)
  const int c0 = (wave & 3) * 16;
  v8f acc = {};
  for (int k0 = 0; k0 < 256; k0 += 4) {
    const int ka = k0 + kh * 2;
    v2f a, b;
    a.x = X[r0 + m][ka];
    a.y = X[r0 + m][ka + 1];
    b.x = W[ka][c0 + m];
    b.y = W[ka + 1][c0 + m];
    acc = wmma4(a, b, acc);
  }
  const int col = c0 + m;
  const float bias = pb[col];
  for (int r = 0; r < 8; ++r) {
    const int row = base + r0 + r + 8 * kh;
    if (row < n) {
      float v = lrelu(acc[r] + bias);
      v += node_emb[(size_t)(nid[row] + 1) * FDIM + col];
      h0[(size_t)row * FDIM + col] = v;
    }
  }
}

// ------------------------------------------------- layer0 fused MLP
// X=[h | (agg-h)/w] (32x128) -> Z=lrelu(X@W1+b1) (32x256) -> O=lrelu(Z@W2+b2)
// h1 = l2norm(O); hres = l2norm(X[:,64:128]).  W2 staged via TDM.
__global__ __launch_bounds__(256) void mlp0_kernel(
    const float* __restrict__ h0, const float* __restrict__ agg0,
    const float* __restrict__ deg, const float* __restrict__ w1,
    const float* __restrict__ b1, const float* __restrict__ w2,
    const float* __restrict__ b2, float* __restrict__ h1,
    float* __restrict__ hres, int n) {
  __shared__ float X[32][129];   // padded
  __shared__ float Z[32][257];   // padded
  __shared__ float O[32][64];
  __shared__ float W2[256][65];  // padded, filled by TDM
  const int tid = threadIdx.x;
  const int base = blockIdx.x * 32;

#if HAVE_TDM
  if (tid < 32) tdm_load_2d_padded(w2, &W2[0][0], 64, 256);
#endif
  for (int i = tid; i < 32 * 128; i += 256) {
    const int row = i >> 7, col = i & 127;
    const int grow = min(base + row, n - 1);
    float v;
    if (col < 64) {
      v = h0[(size_t)grow * FDIM + col];
    } else {
      const int c = col - 64;
      const float w = fmaxf(deg[grow] - 1.f, 1.f);
      v = (agg0[(size_t)grow * FDIM + c] - h0[(size_t)grow * FDIM + c]) / w;
    }
    X[row][col] = v;
  }
#if !HAVE_TDM
  for (int i = tid; i < 256 * 64; i += 256) W2[i >> 6][i & 63] = w2[i];
#endif
  __syncthreads();

  const int lane = tid & 31, wave = tid >> 5;
  const int m = lane & 15, kh = lane >> 4;

  // GEMM1: 2x16 = 32 tiles, 4 per wave, K=128 (B from global; weights L2-hot)
  for (int ti = 0; ti < 4; ++ti) {
    const int t = wave * 4 + ti;
    const int r0 = (t >> 4) * 16, c0 = (t & 15) * 16;
    v8f acc = {};
    for (int k0 = 0; k0 < 128; k0 += 4) {
      const int ka = k0 + kh * 2;
      v2f a, b;
      a.x = X[r0 + m][ka];
      a.y = X[r0 + m][ka + 1];
      b.x = w1[ka * 256 + c0 + m];
      b.y = w1[(ka + 1) * 256 + c0 + m];
      acc = wmma4(a, b, acc);
    }
    const float bias = b1[c0 + m];
    for (int r = 0; r < 8; ++r) Z[r0 + r + 8 * kh][c0 + m] = lrelu(acc[r] + bias);
  }
#if HAVE_TDM
  if (tid < 32) __builtin_amdgcn_s_wait_tensorcnt(0);
#endif
  __syncthreads();

  // GEMM2: 2x4 = 8 tiles, 1 per wave, K=256 (B from LDS W2)
  {
    const int r0 = (wave >> 2) * 16, c0 = (wave & 3) * 16;
    v8f acc = {};
    for (int k0 = 0; k0 < 256; k0 += 4) {
      const int ka = k0 + kh * 2;
      v2f a, b;
      a.x = Z[r0 + m][ka];
      a.y = Z[r0 + m][ka + 1];
      b.x = W2[ka][c0 + m];
      b.y = W2[ka + 1][c0 + m];
      acc = wmma4(a, b, acc);
    }
    const float bias = b2[c0 + m];
    for (int r = 0; r < 8; ++r) O[r0 + r + 8 * kh][c0 + m] = lrelu(acc[r] + bias);
  }
  __syncthreads();

  if (tid < 32) {
    const int grow = base + tid;
    if (grow < n) {
      float ss = 0.f;
      for (int c = 0; c < FDIM; ++c) { const float v = O[tid][c]; ss += v * v; }
      float inv = 1.f / fmaxf(sqrtf(ss), 1e-6f);
      for (int c = 0; c < FDIM; ++c) h1[(size_t)grow * FDIM + c] = O[tid][c] * inv;
      ss = 0.f;
      for (int c = 0; c < FDIM; ++c) { const float v = X[tid][64 + c]; ss += v * v; }
      inv = 1.f / fmaxf(sqrtf(ss), 1e-6f);
      for (int c = 0; c < FDIM; ++c)
        hres[(size_t)grow * FDIM + c] = X[tid][64 + c] * inv;
    }
  }
}

// ------------------------------------------------- layer1 fused MLP
// X=[h | (agg-h)/w | resum] (32x192) -> Z=lrelu(X@W1+b1) (32x384)
// out = l2norm(Z@W2 + b2)   (no trailing lrelu)
__global__ __launch_bounds__(256) void mlp1_kernel(
    const float* __restrict__ h1, const float* __restrict__ agg1,
    const float* __restrict__ resum, const float* __restrict__ deg,
    const float* __restrict__ w1, const float* __restrict__ b1,
    const float* __restrict__ w2, const float* __restrict__ b2,
    float* __restrict__ out, int n) {
  __shared__ float X[32][193];  // padded
  __shared__ float Z[32][385];  // padded
  __shared__ float O[32][64];
  const int tid = threadIdx.x;
  const int base = blockIdx.x * 32;

  for (int i = tid; i < 32 * 192; i += 256) {
    const int row = i / 192, col = i % 192;
    const int grow = min(base + row, n - 1);
    float v;
    if (col < 64) {
      v = h1[(size_t)grow * FDIM + col];
    } else if (col < 128) {
      const int c = col - 64;
      const float w = fmaxf(deg[grow] - 1.f, 1.f);
      v = (agg1[(size_t)grow * FDIM + c] - h1[(size_t)grow * FDIM + c]) / w;
    } else {
      v = resum[(size_t)grow * FDIM + (col - 128)];
    }
    X[row][col] = v;
  }
  __syncthreads();

  const int lane = tid & 31, wave = tid >> 5;
  const int m = lane & 15, kh = lane >> 4;

  // GEMM1: 2x24 = 48 tiles, 6 per wave, K=192
  for (int ti = 0; ti < 6; ++ti) {
    const int t = wave * 6 + ti;
    const int r0 = (t / 24) * 16, c0 = (t % 24) * 16;
    v8f acc = {};
    for (int k0 = 0; k0 < 192; k0 += 4) {
      const int ka = k0 + kh * 2;
      v2f a, b;
      a.x = X[r0 + m][ka];
      a.y = X[r0 + m][ka + 1];
      b.x = w1[ka * 384 + c0 + m];
      b.y = w1[(ka + 1) * 384 + c0 + m];
      acc = wmma4(a, b, acc);
    }
    const float bias = b1[c0 + m];
    for (int r = 0; r < 8; ++r) Z[r0 + r + 8 * kh][c0 + m] = lrelu(acc[r] + bias);
  }
  __syncthreads();

  // GEMM2: 2x4 = 8 tiles, 1 per wave, K=384
  {
    const int r0 = (wave >> 2) * 16, c0 = (wave & 3) * 16;
    v8f acc = {};
    for (int k0 = 0; k0 < 384; k0 += 4) {
      const int ka = k0 + kh * 2;
      v2f a, b;
      a.x = Z[r0 + m][ka];
      a.y = Z[r0 + m][ka + 1];
      b.x = w2[ka * FDIM + c0 + m];
      b.y = w2[(ka + 1) * FDIM + c0 + m];
      acc = wmma4(a, b, acc);
    }
    const float bias = b2[c0 + m];
    for (int r = 0; r < 8; ++r) O[r0 + r + 8 * kh][c0 + m] = acc[r] + bias;
  }
  __syncthreads();

  if (tid < 32) {
    const int grow = base + tid;
    if (grow < n) {
      float ss = 0.f;
      for (int c = 0; c < FDIM; ++c) { const float v = O[tid][c]; ss += v * v; }
      const float inv = 1.f / fmaxf(sqrtf(ss), 1e-6f);
      for (int c = 0; c < FDIM; ++c)
        out[(size_t)grow * FDIM + c] = O[tid][c] * inv;
    }
  }
}

// ---------------------------------------------------------------- launcher
extern "C" void kernel_launch(void* const* d_in, const int* in_sizes, int n_in,
                              void* d_out, int out_size, void* d_ws,
                              size_t ws_size, hipStream_t stream) {
  const float* content = (const float*)d_in[0];
  const float* node_emb = (const float*)d_in[1];
  const float* proj_w = (const float*)d_in[2];
  const float* proj_b = (const float*)d_in[3];
  const float* c0w1 = (const float*)d_in[4];
  const float* c0b1 = (const float*)d_in[5];
  const float* c0w2 = (const float*)d_in[6];
  const float* c0b2 = (const float*)d_in[7];
  const float* c1w1 = (const float*)d_in[8];
  const float* c1b1 = (const float*)d_in[9];
  const float* c1w2 = (const float*)d_in[10];
  const float* c1b2 = (const float*)d_in[11];
  const int* nid = (const int*)d_in[12];
  const int* src = (const int*)d_in[13];
  const int* dst = (const int*)d_in[14];
  const int n = in_sizes[12];  // N nodes
  const int e = in_sizes[13];  // E edges

  float* ws = (float*)d_ws;
  const size_t NF = (size_t)n * FDIM;
  float* h0 = ws;
  float* h1 = ws + NF;
  float* hres = ws + 2 * NF;
  float* agg0 = ws + 3 * NF;  // zeroed accumulators start here
  float* agg1 = ws + 4 * NF;
  float* resum = ws + 5 * NF;
  float* deg = ws + 6 * NF;

  const int zcount = (int)(3 * NF) + n;  // agg0, agg1, resum, deg
  zero_kernel<<<2048, 256, 0, stream>>>(agg0, zcount);

  deg_kernel<<<(e + 255) / 256, 256, 0, stream>>>(dst, deg, e);

  const int nblk = (n + 31) / 32;
  proj_kernel<<<nblk, 256, 0, stream>>>(content, node_emb, proj_w, proj_b, nid,
                                        h0, n);

  const int sthreads = e * 16;
  scatter_kernel<<<(sthreads + 255) / 256, 256, 0, stream>>>(h0, src, dst, agg0,
                                                             e);

  mlp0_kernel<<<nblk, 256, 0, stream>>>(h0, agg0, deg, c0w1, c0b1, c0w2, c0b2,
                                        h1, hres, n);

  scatter_kernel<<<(sthreads + 255) / 256, 256, 0, stream>>>(h1, src, dst, agg1,
                                                             e);
  scatter_kernel<<<(sthreads + 255) / 256, 256, 0, stream>>>(hres, src, dst,
                                                             resum, e);

  mlp1_kernel<<<nblk, 256, 0, stream>>>(h1, agg1, resum, deg, c1w1, c1b1, c1w2,
                                        c1b2, (float*)d_out, n);
}